// ContrastiveLoss_69887707840613
// MI455X (gfx1250) — compile-verified
//
#include <hip/hip_runtime.h>

typedef __attribute__((ext_vector_type(2))) float v2f;
typedef __attribute__((ext_vector_type(8))) float v8f;

// Problem dims (fixed by setup_inputs)
constexpr int Bb = 8;
constexpr int Cc = 256;
constexpr int HW = 128 * 128;   // 16384
constexpr int Nn = 100;
constexpr int Tt = 6;
constexpr int NT = 7;           // ceil(100/16) n-tiles
constexpr int CT = 16;          // 256/16 c-tiles

// ---------------------------------------------------------------------------
// Kernel 1: pooled[b,n,c] = sum_hw visual[b,c,hw] * (mask[b,n,hw] > 0.5)
// via V_WMMA_F32_16X16X4_F32. Also folds msum[b,n] into the same pass.
// One wave per (b, c-tile, n-tile); 4 waves per block.
// ---------------------------------------------------------------------------
__global__ __launch_bounds__(128) void pooled_wmma_kernel(
    const float* __restrict__ visual, const float* __restrict__ masks,
    float* __restrict__ pooled, float* __restrict__ msum)
{
    const int lane   = threadIdx.x & 31;
    const int waveId = blockIdx.x * 4 + (threadIdx.x >> 5);

    const int b   = waveId / (CT * NT);
    const int rem = waveId % (CT * NT);
    const int ct  = rem / NT;
    const int nt  = rem % NT;

    const int half = lane >> 4;   // 0: K={0,1}, 1: K={2,3}
    const int m    = lane & 15;

    const int c = ct * 16 + m;                 // A-matrix row for this lane
    const int n = nt * 16 + m;                 // B-matrix column for this lane
    const bool nvalid = (n < Nn);
    const int n_eff = nvalid ? n : (Nn - 1);   // clamp to keep loads in-bounds

    const float* vptr = visual + (size_t)(b * Cc + c)    * HW + 2 * half;
    const float* mptr = masks  + (size_t)(b * Nn + n_eff) * HW + 2 * half;

    v8f acc = {};
    float ms = 0.0f;

    #pragma unroll 4
    for (int k = 0; k < HW; k += 4) {
        v2f a  = *(const v2f*)(vptr + k);
        v2f mv = *(const v2f*)(mptr + k);
        v2f bb;
        bb.x = (nvalid && (mv.x > 0.5f)) ? 1.0f : 0.0f;
        bb.y = (nvalid && (mv.y > 0.5f)) ? 1.0f : 0.0f;
        ms += bb.x + bb.y;
        // D = A(16x4 f32) * B(4x16 f32) + C   -> v_wmma_f32_16x16x4_f32
        acc = __builtin_amdgcn_wmma_f32_16x16x4_f32(
            /*neg_a=*/false, a, /*neg_b=*/false, bb,
            /*c_mod=*/(short)0, acc, /*reuse_a=*/false, /*reuse_b=*/false);
    }

    // D layout: lane col = m, VGPR j holds row j + 8*half  (32-bit 16x16 C/D)
    if (nvalid) {
        float* out = pooled + (size_t)(b * Nn + n) * Cc + ct * 16 + 8 * half;
        #pragma unroll
        for (int j = 0; j < 8; ++j) out[j] = acc[j];
    }

    // msum: lane m (K%4 in {0,1}) + lane m+16 (K%4 in {2,3}) cover all K
    ms += __shfl_xor(ms, 16);
    if (ct == 0 && half == 0 && nvalid) {
        msum[b * Nn + n] = ms;
    }
}

// ---------------------------------------------------------------------------
// Kernel 2: per-(b,n) normalize + cosine sims vs 6 text rows + CE.
// One wave per item; butterfly reductions keep everything in registers.
// ---------------------------------------------------------------------------
__global__ __launch_bounds__(32) void score_kernel(
    const float* __restrict__ pooled, const float* __restrict__ msum,
    const float* __restrict__ text, const int* __restrict__ labels,
    const float* __restrict__ temp,
    float* __restrict__ cev, float* __restrict__ vld)
{
    const int i    = blockIdx.x;            // 0 .. B*N-1
    const int lane = threadIdx.x;

    const float ms    = msum[i];
    const float denom = fmaxf(ms, 1.0f);

    float p[8];
    float pp = 0.0f;
    #pragma unroll
    for (int j = 0; j < 8; ++j) {
        p[j] = pooled[(size_t)i * Cc + lane + 32 * j] / denom;
        pp += p[j] * p[j];
    }
    #pragma unroll
    for (int off = 16; off >= 1; off >>= 1) pp += __shfl_xor(pp, off);

    const float inv_pn = 1.0f / fmaxf(sqrtf(pp), 1e-12f);
    const float invt   = 1.0f / fabsf(temp[0]);

    float sims[Tt];
    #pragma unroll
    for (int t = 0; t < Tt; ++t) {
        float dot = 0.0f, tn = 0.0f;
        #pragma unroll
        for (int j = 0; j < 8; ++j) {
            float tv = text[t * Cc + lane + 32 * j];
            dot += p[j] * tv;
            tn  += tv * tv;
        }
        #pragma unroll
        for (int off = 16; off >= 1; off >>= 1) {
            dot += __shfl_xor(dot, off);
            tn  += __shfl_xor(tn, off);
        }
        sims[t] = dot * inv_pn * (1.0f / fmaxf(sqrtf(tn), 1e-12f)) * invt;
    }

    if (lane == 0) {
        float mx = sims[0];
        #pragma unroll
        for (int t = 1; t < Tt; ++t) mx = fmaxf(mx, sims[t]);
        float se = 0.0f;
        #pragma unroll
        for (int t = 0; t < Tt; ++t) se += __expf(sims[t] - mx);
        const float lse = mx + __logf(se);

        const int lab = labels[i];
        int tgt = lab - 1;
        tgt = tgt < 0 ? 0 : (tgt > Tt - 1 ? Tt - 1 : tgt);
        const float ce = lse - sims[tgt];
        const float valid = (lab >= 1 && lab <= Tt && ms >= 1.0f) ? 1.0f : 0.0f;
        cev[i] = ce * valid;
        vld[i] = valid;
    }
}

// ---------------------------------------------------------------------------
// Kernel 3: final masked mean over the 800 items.
// ---------------------------------------------------------------------------
__global__ __launch_bounds__(32) void final_reduce_kernel(
    const float* __restrict__ cev, const float* __restrict__ vld,
    float* __restrict__ out)
{
    const int lane = threadIdx.x;
    float s = 0.0f, c = 0.0f;
    for (int i = lane; i < Bb * Nn; i += 32) { s += cev[i]; c += vld[i]; }
    #pragma unroll
    for (int off = 16; off >= 1; off >>= 1) {
        s += __shfl_xor(s, off);
        c += __shfl_xor(c, off);
    }
    if (lane == 0) out[0] = (c > 0.0f) ? (s / fmaxf(c, 1.0f)) : 0.0f;
}

// ---------------------------------------------------------------------------
extern "C" void kernel_launch(void* const* d_in, const int* in_sizes, int n_in,
                              void* d_out, int out_size, void* d_ws, size_t ws_size,
                              hipStream_t stream)
{
    const float* visual = (const float*)d_in[0];  // (B,C,H,W) f32
    const float* text   = (const float*)d_in[1];  // (T,C)     f32
    const float* masks  = (const float*)d_in[2];  // (B,N,H,W) f32
    const int*   labels = (const int*)  d_in[3];  // (B,N)     i32
    const float* temp   = (const float*)d_in[4];  // scalar    f32
    float*       out    = (float*)d_out;

    float* ws     = (float*)d_ws;
    float* pooled = ws;                          // B*N*C  = 204800 floats
    float* msum   = pooled + (size_t)Bb * Nn * Cc; // 800 floats
    float* cev    = msum + Bb * Nn;              // 800 floats
    float* vld    = cev + Bb * Nn;               // 800 floats

    // 896 waves total (8 b * 16 c-tiles * 7 n-tiles), 4 waves per block
    pooled_wmma_kernel<<<(Bb * CT * NT) / 4, 128, 0, stream>>>(visual, masks, pooled, msum);
    score_kernel<<<Bb * Nn, 32, 0, stream>>>(pooled, msum, text, labels, temp, cev, vld);
    final_reduce_kernel<<<1, 32, 0, stream>>>(cev, vld, out);
}